// BeamSearchAttentionDecoder_34651796144639
// MI455X (gfx1250) — compile-verified
//
#include <hip/hip_runtime.h>
#include <math.h>

constexpr int NB = 32;      // batch (beams)
constexpr int NS = 2048;    // sequence
constexpr int NH = 1024;    // hidden
constexpr int NK = 4;       // topk
constexpr int NR = 8;       // NB / NK
constexpr int NPREV = 2;

typedef float v2f __attribute__((ext_vector_type(2)));
typedef float v8f __attribute__((ext_vector_type(8)));

static __device__ __forceinline__ v8f wmma4(v2f a, v2f b, v8f c) {
  // V_WMMA_F32_16X16X4_F32 : D = A(16x4) x B(4x16) + C(16x16), all f32
  return __builtin_amdgcn_wmma_f32_16x16x4_f32(
      /*neg_a=*/false, a, /*neg_b=*/false, b,
      /*c_mod=*/(short)0, c, /*reuse_a=*/false, /*reuse_b=*/false);
}

// ---------------------------------------------------------------------------
// GRU step: gi = x@Wih^T + bih ; gh = h@Whh^T + bhh ; combine.
// 64 blocks (one per 16-column tile of H), 384 threads = 12 waves.
// ---------------------------------------------------------------------------
__global__ void __launch_bounds__(384)
k_gru(const float* __restrict__ x, const float* __restrict__ h,
      const float* __restrict__ Wih, const float* __restrict__ Whh,
      const float* __restrict__ bih, const float* __restrict__ bhh,
      float* __restrict__ hnew)
{
  __shared__ float tiles[12][16][16];
  const int j0   = blockIdx.x * 16;
  const int tid  = threadIdx.x;
  const int w    = tid >> 5;
  const int lane = tid & 31;

  const int mat  = w / 6;          // 0: (x, Wih)   1: (h, Whh)
  const int gate = (w % 6) >> 1;   // 0: r  1: z  2: n
  const int mt   = w & 1;          // batch-row tile

  const float* A = mat ? h : x;
  const float* W = mat ? Whh : Wih;
  const int row0 = mt * 16;
  const int col0 = gate * NH + j0;

  const int lrow = lane & 15;
  const int koff = (lane >> 4) << 1;
  const float* arow = A + (size_t)(row0 + lrow) * NH + koff;
  const float* wrow = W + (size_t)(col0 + lrow) * NH + koff;

  v8f acc = {};
  for (int kk = 0; kk < NH; kk += 4) {
    v2f a  = *(const v2f*)(arow + kk);
    v2f bm = *(const v2f*)(wrow + kk);
    acc = wmma4(a, bm, acc);
  }
  {
    const int mbase = (lane >> 4) << 3;
    const int n = lane & 15;
#pragma unroll
    for (int v = 0; v < 8; ++v) tiles[w][mbase + v][n] = acc[v];
  }
  __syncthreads();

  for (int idx = tid; idx < 32 * 16; idx += 384) {
    const int m = idx >> 4, n = idx & 15;
    const int mt2 = m >> 4, mm = m & 15;
    const int j = j0 + n;
    const float ir = tiles[0  + mt2][mm][n] + bih[j];
    const float iz = tiles[2  + mt2][mm][n] + bih[NH + j];
    const float in = tiles[4  + mt2][mm][n] + bih[2*NH + j];
    const float hr = tiles[6  + mt2][mm][n] + bhh[j];
    const float hz = tiles[8  + mt2][mm][n] + bhh[NH + j];
    const float hn = tiles[10 + mt2][mm][n] + bhh[2*NH + j];
    const float r  = 1.f / (1.f + expf(-(ir + hr)));
    const float z  = 1.f / (1.f + expf(-(iz + hz)));
    const float nn = tanhf(in + r * hn);
    hnew[m * NH + j] = (1.f - z) * nn + z * h[m * NH + j];
  }
}

// ---------------------------------------------------------------------------
// Generic 32x1024 = (32x1024)x(1024x1024) WMMA GEMM.
//  transposedW==1 : Out = A @ W        -- q = hnew@Wk
//  transposedW==0 : Out = A @ W^T + b  -- context = ctxpre@Wv^T + bv
// ---------------------------------------------------------------------------
__global__ void __launch_bounds__(128)
k_gemm32(const float* __restrict__ A, const float* __restrict__ W,
         const float* __restrict__ bias, float* __restrict__ Out, int transposedW)
{
  const int tid  = threadIdx.x;
  const int w    = tid >> 5;
  const int lane = tid & 31;
  const int t    = blockIdx.x * 4 + w;   // 0..127
  const int mt   = t & 1;
  const int n0   = (t >> 1) << 4;
  const int row0 = mt * 16;
  const int lrow = lane & 15;
  const int koff = (lane >> 4) << 1;
  const float* arow = A + (size_t)(row0 + lrow) * NH + koff;

  v8f acc = {};
  if (transposedW) {
    const float* wcol = W + n0 + lrow;
    for (int kk = 0; kk < NH; kk += 4) {
      v2f a = *(const v2f*)(arow + kk);
      v2f bm;
      bm.x = wcol[(size_t)(kk + koff) * NH];
      bm.y = wcol[(size_t)(kk + koff + 1) * NH];
      acc = wmma4(a, bm, acc);
    }
  } else {
    const float* wrow = W + (size_t)(n0 + lrow) * NH + koff;
    for (int kk = 0; kk < NH; kk += 4) {
      v2f a  = *(const v2f*)(arow + kk);
      v2f bm = *(const v2f*)(wrow + kk);
      acc = wmma4(a, bm, acc);
    }
  }
  const int mbase = (lane >> 4) << 3;
  const int n = lane & 15;
  const float bb = bias ? bias[n0 + n] : 0.f;
#pragma unroll
  for (int v = 0; v < 8; ++v)
    Out[(size_t)(row0 + mbase + v) * NH + n0 + n] = acc[v] + bb;
}

// ---------------------------------------------------------------------------
// hbk[b] = hnew[b] . bk   (one block, wave w owns beam w)
// ---------------------------------------------------------------------------
__global__ void __launch_bounds__(1024)
k_hbk(const float* __restrict__ hnew, const float* __restrict__ bk,
      float* __restrict__ hbk)
{
  const int w = threadIdx.x >> 5, lane = threadIdx.x & 31;
  float acc = 0.f;
  for (int j = lane; j < NH; j += 32) acc += hnew[w * NH + j] * bk[j];
  for (int o = 16; o; o >>= 1) acc += __shfl_xor(acc, o, 32);
  if (lane == 0) hbk[w] = acc;
}

// ---------------------------------------------------------------------------
// Logits: grid (NB, 8 s-chunks). Wave handles 32 consecutive rows; coalesced
// float4 dot against q staged in LDS; prefetch one row ahead.
// ---------------------------------------------------------------------------
__global__ void __launch_bounds__(256)
k_logits(const float* __restrict__ eo, const float* __restrict__ q,
         const float* __restrict__ hbk, const float* __restrict__ mask,
         float* __restrict__ logits)
{
  __shared__ float s_q[NH];
  const int b     = blockIdx.x;
  const int chunk = blockIdx.y;            // 0..7
  const int tid = threadIdx.x, lane = tid & 31, w = tid >> 5;
  for (int j = tid; j < NH; j += 256) s_q[j] = q[b * NH + j];
  __syncthreads();
  const float cb = hbk[b];
  const float4* q4  = (const float4*)s_q;
  const float4* eob = (const float4*)(eo + (size_t)b * NS * NH);
  const int s0 = chunk * 256 + w * 32;
  for (int i = 0; i < 32; ++i) {
    const int s = s0 + i;
    const float4* row = eob + (size_t)s * (NH / 4);
    if (i + 1 < 32)
      __builtin_prefetch((const void*)(row + NH / 4 + lane * 8), 0, 0);
    float acc = 0.f;
#pragma unroll
    for (int t = 0; t < 8; ++t) {
      const float4 e  = row[lane + t * 32];
      const float4 qv = q4[lane + t * 32];
      acc += e.x * qv.x + e.y * qv.y + e.z * qv.z + e.w * qv.w;
    }
    for (int o = 16; o; o >>= 1) acc += __shfl_xor(acc, o, 32);
    if (lane == 0)
      logits[b * NS + s] = (acc + cb) * (1.f / 32.f) + mask[b * NS + s];
  }
}

// ---------------------------------------------------------------------------
// Softmax + top-4 per beam (32 small blocks; all data in LDS).
// ---------------------------------------------------------------------------
__global__ void __launch_bounds__(256)
k_softmax_topk(const float* __restrict__ logits, float* __restrict__ alpha,
               float* __restrict__ scores, int* __restrict__ sents)
{
  __shared__ float s_a[NS];
  __shared__ float red[256];
  __shared__ int   redi[256];
  __shared__ float s_max, s_sum;
  const int b = blockIdx.x, tid = threadIdx.x;

  for (int s = tid; s < NS; s += 256) s_a[s] = logits[b * NS + s];
  __syncthreads();

  float mv = -INFINITY;
  for (int s = tid; s < NS; s += 256) mv = fmaxf(mv, s_a[s]);
  red[tid] = mv; __syncthreads();
  for (int o = 128; o; o >>= 1) { if (tid < o) red[tid] = fmaxf(red[tid], red[tid + o]); __syncthreads(); }
  if (tid == 0) s_max = red[0];
  __syncthreads();
  const float mx = s_max;
  float sm = 0.f;
  for (int s = tid; s < NS; s += 256) sm += expf(s_a[s] - mx);
  red[tid] = sm; __syncthreads();
  for (int o = 128; o; o >>= 1) { if (tid < o) red[tid] += red[tid + o]; __syncthreads(); }
  if (tid == 0) s_sum = red[0];
  __syncthreads();
  const float inv = 1.f / s_sum;
  for (int s = tid; s < NS; s += 256) {
    const float a = expf(s_a[s] - mx) * inv;
    s_a[s] = a;
    alpha[b * NS + s] = a;
  }
  __syncthreads();

  for (int k = 0; k < NK; ++k) {
    float bv = -1.f; int bi = NS;
    for (int s = tid; s < NS; s += 256) {
      const float v = s_a[s];
      if (v > bv || (v == bv && s < bi)) { bv = v; bi = s; }
    }
    red[tid] = bv; redi[tid] = bi; __syncthreads();
    for (int o = 128; o; o >>= 1) {
      if (tid < o) {
        const float ov = red[tid + o]; const int oi = redi[tid + o];
        if (ov > red[tid] || (ov == red[tid] && oi < redi[tid])) { red[tid] = ov; redi[tid] = oi; }
      }
      __syncthreads();
    }
    if (tid == 0) {
      scores[b * NK + k] = red[0];
      sents [b * NK + k] = redi[0];
      s_a[redi[0]] = -1.f;
    }
    __syncthreads();
  }
}

// ---------------------------------------------------------------------------
// ctx_pre: grid (NB, 8 h-chunks of 128). Waves partition s; 512B coalesced
// reads per wave per row; serial LDS sum of 8 partials per column.
// ---------------------------------------------------------------------------
__global__ void __launch_bounds__(256)
k_ctx(const float* __restrict__ eo, const float* __restrict__ alpha,
      float* __restrict__ ctxpre)
{
  __shared__ float s_alpha[NS];
  __shared__ float part[8][128];
  const int b  = blockIdx.x;
  const int hc = blockIdx.y;               // 0..7 -> h0 = hc*128
  const int tid = threadIdx.x, t = tid & 31, g = tid >> 5;

  for (int s = tid; s < NS; s += 256) s_alpha[s] = alpha[b * NS + s];
  __syncthreads();

  const float4* eob = (const float4*)(eo + (size_t)b * NS * NH) + hc * 32;
  float4 acc = {0.f, 0.f, 0.f, 0.f};
  for (int s = g; s < NS; s += 8) {
    if (s + 8 < NS && t < 4)
      __builtin_prefetch((const void*)(eob + (size_t)(s + 8) * (NH / 4) + t * 8), 0, 0);
    const float a  = s_alpha[s];
    const float4 e = eob[(size_t)s * (NH / 4) + t];
    acc.x += a * e.x; acc.y += a * e.y; acc.z += a * e.z; acc.w += a * e.w;
  }
  part[g][t * 4 + 0] = acc.x;
  part[g][t * 4 + 1] = acc.y;
  part[g][t * 4 + 2] = acc.z;
  part[g][t * 4 + 3] = acc.w;
  __syncthreads();
  if (tid < 128) {
    float sum = 0.f;
#pragma unroll
    for (int gg = 0; gg < 8; ++gg) sum += part[gg][tid];
    ctxpre[b * NH + hc * 128 + tid] = sum;
  }
}

// ---------------------------------------------------------------------------
// Beam combine (unchanged).
// ---------------------------------------------------------------------------
__global__ void k_combine(const float* __restrict__ scores, const int* __restrict__ sents,
                          const float* __restrict__ evidence, const int* __restrict__ evidx,
                          float* __restrict__ out_idx, float* __restrict__ out_ev,
                          int* __restrict__ sflat)
{
  const int r = threadIdx.x;
  if (r >= NR) return;
  float vals[NK * NK]; int ord[NK * NK];
  for (int kb = 0; kb < NK; ++kb)
    for (int kk = 0; kk < NK; ++kk) {
      const int bsrc = r * NK + kb;
      const int f = kb * NK + kk;
      vals[f] = -logf(scores[bsrc * NK + kk]) + evidence[bsrc];
      ord[f]  = f;
    }
  for (int o = 0; o < NK; ++o) {
    int best = o;
    for (int j = o + 1; j < NK * NK; ++j)
      if (vals[ord[j]] < vals[ord[best]] ||
          (vals[ord[j]] == vals[ord[best]] && ord[j] < ord[best])) best = j;
    const int tmp = ord[o]; ord[o] = ord[best]; ord[best] = tmp;
  }
  for (int o = 0; o < NK; ++o) {
    const int f  = ord[o];
    const int kb = f / NK, kk = f % NK;
    const int bsrc = r * NK + kb;
    const int bout = r * NK + o;
    out_ev[bout] = vals[f];
    sflat[bout]  = bsrc;
    out_idx[bout * 3 + 0] = (float)evidx[bsrc * NPREV + 0];
    out_idx[bout * 3 + 1] = (float)evidx[bsrc * NPREV + 1];
    out_idx[bout * 3 + 2] = (float)sents[bsrc * NK + kk];
  }
}

// ---------------------------------------------------------------------------
// Gather / assemble remaining outputs (unchanged).
// ---------------------------------------------------------------------------
__global__ void __launch_bounds__(256)
k_gather(const float* __restrict__ context, const float* __restrict__ hnew,
         const float* __restrict__ logits, const float* __restrict__ attn_prev,
         const float* __restrict__ mask, const int* __restrict__ sflat,
         float* __restrict__ out)
{
  constexpr int RES = NB * NH;
  constexpr int HID = NB * NH;
  constexpr int ATT = 3 * NB * NS;
  constexpr int MSK = NB * NS;
  constexpr int off1 = RES;
  constexpr int off2 = off1 + HID;
  constexpr int off3 = off2 + NB * 3;
  constexpr int off4 = off3 + ATT;
  const long i = (long)blockIdx.x * blockDim.x + threadIdx.x;
  if (i < RES) {
    const int b = (int)(i / NH), hh = (int)(i % NH);
    out[i] = context[sflat[b] * NH + hh];
  } else if (i < RES + HID) {
    const long j = i - RES; const int b = (int)(j / NH), hh = (int)(j % NH);
    out[off1 + j] = hnew[sflat[b] * NH + hh];
  } else if (i < RES + HID + ATT) {
    const long j = i - RES - HID;
    const int p = (int)(j / (NB * NS));
    const long rem = j % (NB * NS);
    const int b = (int)(rem / NS), s = (int)(rem % NS);
    const float v = (p < 2) ? attn_prev[(long)p * NB * NS + (long)sflat[b] * NS + s]
                            : logits[(long)sflat[b] * NS + s];
    out[off3 + j] = v;
  } else if (i < RES + HID + ATT + MSK) {
    const long j = i - RES - HID - ATT;
    const int b = (int)(j / NS), s = (int)(j % NS);
    out[off4 + j] = mask[(long)sflat[b] * NS + s];
  }
}

// ---------------------------------------------------------------------------
extern "C" void kernel_launch(void* const* d_in, const int* in_sizes, int n_in,
                              void* d_out, int out_size, void* d_ws, size_t ws_size,
                              hipStream_t stream)
{
  const float* last_hidden = (const float*)d_in[0];
  const float* dec_in      = (const float*)d_in[1];
  const float* eo          = (const float*)d_in[2];
  const float* attn_prev   = (const float*)d_in[3];
  const float* mask        = (const float*)d_in[4];
  const float* evidence    = (const float*)d_in[5];
  const int*   evidx       = (const int*)d_in[6];
  const float* Wih         = (const float*)d_in[7];
  const float* Whh         = (const float*)d_in[8];
  const float* bih         = (const float*)d_in[9];
  const float* bhh         = (const float*)d_in[10];
  const float* Wk          = (const float*)d_in[11];
  const float* bk          = (const float*)d_in[12];
  const float* Wv          = (const float*)d_in[13];
  const float* bv          = (const float*)d_in[14];

  float* wsF     = (float*)d_ws;
  float* hnew    = wsF;               // 32768
  float* q       = wsF + 32768;       // 32768
  float* logits  = wsF + 65536;       // 65536
  float* alpha   = wsF + 131072;      // 65536
  float* ctxpre  = wsF + 196608;      // 32768
  float* context = wsF + 229376;      // 32768
  float* scores  = wsF + 262144;      // 128
  float* hbk     = wsF + 262272;      // 32
  int*   wsI     = (int*)(wsF + 262304);
  int*   sents   = wsI;               // 128
  int*   sflat   = wsI + 128;         // 32

  float* out = (float*)d_out;
  constexpr int off2 = 2 * NB * NH;
  constexpr int off5 = off2 + NB * 3 + 3 * NB * NS + NB * NS;

  k_gru         <<<64,            384, 0, stream>>>(dec_in, last_hidden, Wih, Whh, bih, bhh, hnew);
  k_hbk         <<<1,            1024, 0, stream>>>(hnew, bk, hbk);
  k_gemm32      <<<32,            128, 0, stream>>>(hnew, Wk, nullptr, q, /*transposedW=*/1);
  k_logits      <<<dim3(NB, 8),   256, 0, stream>>>(eo, q, hbk, mask, logits);
  k_softmax_topk<<<NB,            256, 0, stream>>>(logits, alpha, scores, sents);
  k_ctx         <<<dim3(NB, 8),   256, 0, stream>>>(eo, alpha, ctxpre);
  k_gemm32      <<<32,            128, 0, stream>>>(ctxpre, Wv, bv, context, /*transposedW=*/0);
  k_combine     <<<1,               8, 0, stream>>>(scores, sents, evidence, evidx, out + off2, out + off5, sflat);
  k_gather      <<<1280,          256, 0, stream>>>(context, hnew, logits, attn_prev, mask, sflat, out);
}